// ChebConvNet_82463372083215
// MI455X (gfx1250) — compile-verified
//
#include <hip/hip_runtime.h>

// Problem constants (fixed by the reference setup)
#define B_GRAPHS 128
#define F_NODES  2000
#define C_HID    32
#define N_NODES  (B_GRAPHS * F_NODES)   // 256000
#define E_EDGES  (N_NODES * 16)         // 4096000

typedef __attribute__((ext_vector_type(2))) float v2f;
typedef __attribute__((ext_vector_type(8))) float v8f;

// ---------------------------------------------------------------- utilities
__global__ __launch_bounds__(256) void fill_zero(float* __restrict__ p, int n) {
  int i = blockIdx.x * blockDim.x + threadIdx.x;
  if (i < n) p[i] = 0.0f;
}

__global__ __launch_bounds__(256) void neg_copy(const float* __restrict__ a,
                                                float* __restrict__ o, int n) {
  int i = blockIdx.x * blockDim.x + threadIdx.x;
  if (i < n) o[i] = -a[i];
}

// deg[src[e]] += 1  (hardware f32 atomics; deg array is 1MB -> L2 resident)
__global__ __launch_bounds__(256) void deg_count(const int* __restrict__ src,
                                                 float* __restrict__ deg, int e) {
  int i = blockIdx.x * blockDim.x + threadIdx.x;
  if (i < e) unsafeAtomicAdd(&deg[src[i]], 1.0f);
}

// in-place: dis = deg > 0 ? rsqrt(deg) : 0
__global__ __launch_bounds__(256) void deg_to_dis(float* __restrict__ p, int n) {
  int i = blockIdx.x * blockDim.x + threadIdx.x;
  if (i < n) {
    float d = p[i];
    p[i] = (d > 0.0f) ? rsqrtf(d) : 0.0f;
  }
}

// hout[dst] += coef * (-dis[src]*dis[dst]) * hin[src]
// norm is recomputed from the 1MB L2-resident dis array instead of streaming
// a 16MB norm buffer 4x from HBM.
__global__ __launch_bounds__(256) void edge_scatter(const int* __restrict__ src,
                                                    const int* __restrict__ dst,
                                                    const float* __restrict__ dis,
                                                    const float* __restrict__ hin,
                                                    float* __restrict__ hout,
                                                    float coef, int e) {
  int i = blockIdx.x * blockDim.x + threadIdx.x;
  if (i < e) {
    int s = src[i];
    int d = dst[i];
    float v = -coef * dis[s] * dis[d] * hin[s];
    unsafeAtomicAdd(&hout[d], v);
  }
}

// ------------------------------------------------------------- fused head
// Per graph (one workgroup of 8 waves): for each 16-node tile compute
//   h[16,32] = relu( [T0..T4 | 0-pad][16x8] @ W_pad[8x32] + b )  via WMMA f32 16x16x4,
// then contract elementwise with lin_w[(f*32+c)*2 + j] into per-thread partial
// sums, LDS tree-reduce, and emit out[g, 0:2].
__global__ __launch_bounds__(256)
void cheb_head(const float* __restrict__ T0, const float* __restrict__ T1,
               const float* __restrict__ T2, const float* __restrict__ T3,
               const float* __restrict__ T4,
               const float* __restrict__ Wc,   // [5,32]  (K, C_HID), C_IN=1
               const float* __restrict__ bc,   // [32]
               const float* __restrict__ lw,   // [F*32, 2]
               const float* __restrict__ lb,   // [2]
               float* __restrict__ out) {      // [B, 2]
  __shared__ float r0[256];
  __shared__ float r1[256];

  const int g    = blockIdx.x;
  const int tid  = threadIdx.x;
  const int wave = tid >> 5;
  const int lane = tid & 31;
  const int half = lane >> 4;   // 0: lanes 0-15, 1: lanes 16-31
  const int l16  = lane & 15;

  // B fragments (4x16 f32 per WMMA): VGPR0 holds rows K0(lo lanes)/K2(hi lanes),
  // VGPR1 holds K1/K3.  Second WMMA covers K=4..7 (only K=4 nonzero).
  v2f  bfrag0[2], bfrag1[2];
  float bias[2];
#pragma unroll
  for (int nc = 0; nc < 2; ++nc) {
    int col = nc * 16 + l16;
    bfrag0[nc].x = Wc[(half ? 2 : 0) * C_HID + col];
    bfrag0[nc].y = Wc[(half ? 3 : 1) * C_HID + col];
    bfrag1[nc].x = half ? 0.0f : Wc[4 * C_HID + col];
    bfrag1[nc].y = 0.0f;
    bias[nc] = bc[col];
  }

  float s0 = 0.0f, s1 = 0.0f;
  const int base_node = g * F_NODES;

  // F_NODES/16 = 125 tiles; waves stride over tiles (wave-uniform loop, EXEC all-1s)
  for (int tile = wave; tile < F_NODES / 16; tile += 8) {
    const int node = base_node + tile * 16 + l16;  // A: both lane halves load M = l16

    // A fragment (16x4 f32): VGPR0 = K0(lo)/K2(hi), VGPR1 = K1/K3.
    v2f a0, a1;
    a0.x = half ? T2[node] : T0[node];
    a0.y = half ? T3[node] : T1[node];
    a1.x = half ? 0.0f : T4[node];   // K=4 (lo) / K=6 pad (hi)
    a1.y = 0.0f;                     // K=5 / K=7 pad

#pragma unroll
    for (int nc = 0; nc < 2; ++nc) {
      v8f acc = {};
      acc = __builtin_amdgcn_wmma_f32_16x16x4_f32(
          false, a0, false, bfrag0[nc], (short)0, acc, false, false);
      acc = __builtin_amdgcn_wmma_f32_16x16x4_f32(
          false, a1, false, bfrag1[nc], (short)0, acc, false, false);

      const int col = nc * 16 + l16;  // channel index of this lane's column
#pragma unroll
      for (int v = 0; v < 8; ++v) {
        // C/D layout: element (M = v + 8*half, N = l16)
        float h = acc[v] + bias[nc];
        h = (h > 0.0f) ? h : 0.0f;
        const int f    = tile * 16 + v + (half ? 8 : 0);
        const int base = (f * C_HID + col) * 2;   // lin_w row, 2 classes
        s0 = fmaf(h, lw[base],     s0);
        s1 = fmaf(h, lw[base + 1], s1);
      }
    }
  }

  r0[tid] = s0;
  r1[tid] = s1;
  __syncthreads();
#pragma unroll
  for (int off = 128; off > 0; off >>= 1) {
    if (tid < off) {
      r0[tid] += r0[tid + off];
      r1[tid] += r1[tid + off];
    }
    __syncthreads();
  }
  if (tid == 0) {
    out[g * 2 + 0] = r0[0] + lb[0];
    out[g * 2 + 1] = r1[0] + lb[1];
  }
}

// ---------------------------------------------------------------- launcher
extern "C" void kernel_launch(void* const* d_in, const int* in_sizes, int n_in,
                              void* d_out, int out_size, void* d_ws, size_t ws_size,
                              hipStream_t stream) {
  (void)in_sizes; (void)n_in; (void)out_size; (void)ws_size;

  const float* x  = (const float*)d_in[0];           // [N,1]
  const int*   ei = (const int*)d_in[1];             // [2,E]
  /* d_in[2] = batch, unused (graph id derived from node index) */
  const float* Wc = (const float*)d_in[3];           // [5,1,32]
  const float* bc = (const float*)d_in[4];           // [32]
  const float* lw = (const float*)d_in[5];           // [F*32, 2]
  const float* lb = (const float*)d_in[6];           // [2]
  float* out = (float*)d_out;                        // [B,2]

  const int* src = ei;
  const int* dst = ei + E_EDGES;

  // Workspace: dis + T1..T4, each N floats (5 * 1MB = 5.1 MB)
  float* dis = (float*)d_ws;
  float* Tn1 = dis + N_NODES;
  float* Tn2 = Tn1 + N_NODES;
  float* Tn3 = Tn2 + N_NODES;
  float* Tn4 = Tn3 + N_NODES;

  const dim3 blk(256);
  const int gN = (N_NODES + 255) / 256;
  const int gE = (E_EDGES + 255) / 256;

  // degree -> dis = rsqrt(deg)
  fill_zero<<<gN, blk, 0, stream>>>(dis, N_NODES);
  deg_count<<<gE, blk, 0, stream>>>(src, dis, E_EDGES);
  deg_to_dis<<<gN, blk, 0, stream>>>(dis, N_NODES);

  // T1 = L_hat x
  fill_zero<<<gN, blk, 0, stream>>>(Tn1, N_NODES);
  edge_scatter<<<gE, blk, 0, stream>>>(src, dst, dis, x, Tn1, 1.0f, E_EDGES);

  // T_k = 2 L_hat T_{k-1} - T_{k-2}  (init target with -T_{k-2}, scatter 2x)
  neg_copy<<<gN, blk, 0, stream>>>(x, Tn2, N_NODES);
  edge_scatter<<<gE, blk, 0, stream>>>(src, dst, dis, Tn1, Tn2, 2.0f, E_EDGES);

  neg_copy<<<gN, blk, 0, stream>>>(Tn1, Tn3, N_NODES);
  edge_scatter<<<gE, blk, 0, stream>>>(src, dst, dis, Tn2, Tn3, 2.0f, E_EDGES);

  neg_copy<<<gN, blk, 0, stream>>>(Tn2, Tn4, N_NODES);
  edge_scatter<<<gE, blk, 0, stream>>>(src, dst, dis, Tn3, Tn4, 2.0f, E_EDGES);

  // Fused WMMA head + per-graph contraction: one workgroup per graph
  cheb_head<<<B_GRAPHS, blk, 0, stream>>>(x, Tn1, Tn2, Tn3, Tn4,
                                          Wc, bc, lw, lb, out);
}